// BarlowTwinsDiversityLoss_78932908966303
// MI455X (gfx1250) — compile-verified
//
#include <hip/hip_runtime.h>

typedef float v2f __attribute__((ext_vector_type(2)));
typedef float v8f __attribute__((ext_vector_type(8)));

#define D_DIM   128
#define K_HEADS 8
#define EPS_F   1e-8f

// One wave (32 lanes) processes TWO nodes:
//   A (16x128): rows 0-7  = heads 0..7 of node (2w)
//               rows 8-15 = heads 0..7 of node (2w+1)
// G = A * A^T accumulated via 32x v_wmma_f32_16x16x4_f32 (pure f32 path).
// Diagonal 8x8 blocks of G are the per-node Gram matrices.
__global__ __launch_bounds__(256) void barlow_gram_kernel(
    const float* __restrict__ h0, const float* __restrict__ h1,
    const float* __restrict__ h2, const float* __restrict__ h3,
    const float* __restrict__ h4, const float* __restrict__ h5,
    const float* __restrict__ h6, const float* __restrict__ h7,
    float* __restrict__ partials, int N)
{
    const int lane  = threadIdx.x & 31;
    const int wave  = threadIdx.x >> 5;
    const int gwave = blockIdx.x * (blockDim.x >> 5) + wave;

    const int nodeA = gwave * 2;
    const int M     = lane & 15;     // A-matrix row fed by this lane
    const int half  = lane >> 4;     // 0 -> K={0,1}; 1 -> K={2,3}
    const int head  = M & 7;
    const int node  = nodeA + (M >> 3);
    const int nodeClamped = (node < N) ? node : (N - 1);

    // Branchless per-lane head pointer selection (v_cndmask chain).
    const float* hp = h0;
    hp = (head == 1) ? h1 : hp;
    hp = (head == 2) ? h2 : hp;
    hp = (head == 3) ? h3 : hp;
    hp = (head == 4) ? h4 : hp;
    hp = (head == 5) ? h5 : hp;
    hp = (head == 6) ? h6 : hp;
    hp = (head == 7) ? h7 : hp;

    const float* row = hp + (size_t)nodeClamped * D_DIM + 2 * half;

    v8f c = {0.f, 0.f, 0.f, 0.f, 0.f, 0.f, 0.f, 0.f};

    // K-loop: 4 columns per WMMA step; lane loads its aligned float2 slice.
    // Same registers as A and B operand => A * A^T (B lane n = column n's K-slice).
    #pragma unroll
    for (int t = 0; t < D_DIM / 4; ++t) {
        v2f a = *(const v2f*)(row + 4 * t);
        c = __builtin_amdgcn_wmma_f32_16x16x4_f32(
                /*neg_a=*/false, a, /*neg_b=*/false, a,
                /*c_mod=*/(short)0, c, /*reuse_a=*/false, /*reuse_b=*/false);
    }

    // C/D layout: VGPR j, lanes 0-15 -> G[j][lane]; lanes 16-31 -> G[j+8][lane-16].
    // Node-a Gram columns live in lanes 0-7 (rows 0-7), node-b in lanes 24-31 (rows 8-15).
    float g[8];
    #pragma unroll
    for (int j = 0; j < 8; ++j) g[j] = c[j];

    const int base = (lane < 16) ? 0 : 24;  // lane holding column k's diagonal
    const int crel = lane & 7;              // this lane's column within its 8x8 block

    float invk[8];
    float invc = 0.f;
    #pragma unroll
    for (int k = 0; k < 8; ++k) {
        float nsq = __shfl(g[k], base + k, 32);      // G[k][k] = ||x_k||^2
        float nk  = fmaxf(sqrtf(nsq), EPS_F);
        invk[k]   = 1.0f / nk;
        if (k == crel) invc = invk[k];
    }

    float loss = 0.f;
    #pragma unroll
    for (int k = 0; k < 8; ++k) {
        float cc = g[k] * invk[k] * invc;            // normalized cross-correlation
        float d  = cc - ((k == crel) ? 1.0f : 0.f);  // subtract identity
        loss += d * d;
    }

    const bool valid = ((lane < 8) || (lane >= 24)) && (node < N);
    loss = valid ? loss : 0.f;

    // Deterministic wave reduction.
    #pragma unroll
    for (int off = 16; off > 0; off >>= 1)
        loss += __shfl_xor(loss, off, 32);

    __shared__ float wsum[8];
    if (lane == 0) wsum[wave] = loss;
    __syncthreads();

    if (threadIdx.x == 0) {
        float s = 0.f;
        const int nwaves = blockDim.x >> 5;
        for (int i = 0; i < nwaves; ++i) s += wsum[i];
        partials[blockIdx.x] = s;
    }
}

__global__ __launch_bounds__(256) void barlow_reduce_kernel(
    const float* __restrict__ partials, int n, float* __restrict__ out, float invN)
{
    float s = 0.f;
    for (int i = threadIdx.x; i < n; i += 256) s += partials[i];

    __shared__ float sm[256];
    sm[threadIdx.x] = s;
    __syncthreads();
    #pragma unroll
    for (int stride = 128; stride > 0; stride >>= 1) {
        if (threadIdx.x < (unsigned)stride) sm[threadIdx.x] += sm[threadIdx.x + stride];
        __syncthreads();
    }
    if (threadIdx.x == 0) out[0] = sm[0] * invN;
}

extern "C" void kernel_launch(void* const* d_in, const int* in_sizes, int n_in,
                              void* d_out, int out_size, void* d_ws, size_t ws_size,
                              hipStream_t stream)
{
    (void)n_in; (void)out_size; (void)ws_size;

    const float* h0 = (const float*)d_in[0];
    const float* h1 = (const float*)d_in[1];
    const float* h2 = (const float*)d_in[2];
    const float* h3 = (const float*)d_in[3];
    const float* h4 = (const float*)d_in[4];
    const float* h5 = (const float*)d_in[5];
    const float* h6 = (const float*)d_in[6];
    const float* h7 = (const float*)d_in[7];

    const int N = in_sizes[0] / D_DIM;           // 100000
    // 256 threads = 8 waves; 2 nodes per wave -> 16 nodes per block.
    const int blocks = (N + 15) / 16;            // 6250

    float* partials = (float*)d_ws;

    barlow_gram_kernel<<<blocks, 256, 0, stream>>>(
        h0, h1, h2, h3, h4, h5, h6, h7, partials, N);

    barlow_reduce_kernel<<<1, 256, 0, stream>>>(
        partials, blocks, (float*)d_out, 1.0f / (float)N);
}